// ProbabilityDropout_15264313770625
// MI455X (gfx1250) — compile-verified
//
#include <hip/hip_runtime.h>
#include <stdint.h>

// Problem constants (fixed by the reference's setup_inputs)
#define KB      128      // batch rows
#define KDIM    1024     // latent dim
#define KNB     16384    // histogram bins == num_outputs
#define KMULT   128      // eps rows per batch row
#define KSPLIT  8        // min/max blocks per batch row
#define KROWS_PER_CHUNK (KMULT / KSPLIT)   // 16 rows per minmax block
#define KGROUPS (KROWS_PER_CHUNK / 4)      // 4 row-groups (4 rows each) per block
#define KSTG    3        // LDS stage slots (16 KB each)
#define KZP     1e-10f

#ifndef __has_builtin
#define __has_builtin(x) 0
#endif

#if defined(__HIP_DEVICE_COMPILE__) && __has_builtin(__builtin_amdgcn_global_load_async_to_lds_b32)
#define USE_ASYNC_LDS 1
#else
#define USE_ASYNC_LDS 0
#endif

#if defined(__HIP_DEVICE_COMPILE__) && __has_builtin(__builtin_amdgcn_global_load_async_to_lds_b128)
#define USE_ASYNC_B128 1
#else
#define USE_ASYNC_B128 0
#endif

#if defined(__HIP_DEVICE_COMPILE__) && __has_builtin(__builtin_amdgcn_s_wait_asynccnt)
#define USE_WAIT_BUILTIN 1
#else
#define USE_WAIT_BUILTIN 0
#endif

// Decisive probe: round 2 compiled with the __has_builtin gate in place; if this
// errors, the builtins were never present and we must switch to inline asm.
#if defined(__HIP_DEVICE_COMPILE__) && defined(__gfx1250__) && !USE_ASYNC_LDS
#error "gfx1250 device pass is missing __builtin_amdgcn_global_load_async_to_lds_b32"
#endif

typedef int v4i __attribute__((ext_vector_type(4)));

template <int N>
__device__ __forceinline__ void wait_asynccnt() {
#if USE_WAIT_BUILTIN
    __builtin_amdgcn_s_wait_asynccnt(N);
#elif defined(__gfx1250__)
    asm volatile("s_wait_asynccnt %0" :: "i"(N));
#endif
}

// Async ops issued per 16-byte stage element (for wait-count scaling)
#if USE_ASYNC_B128
#define KISS 1
#elif USE_ASYNC_LDS
#define KISS 4
#else
#define KISS 1
#endif

// Stage 16 bytes per lane from global into LDS via the async copy engine.
__device__ __forceinline__ void stage_b128(const float4* g, float4* l) {
#if USE_ASYNC_B128
    __builtin_amdgcn_global_load_async_to_lds_b128(
        (__attribute__((address_space(1))) v4i*)g,
        (__attribute__((address_space(3))) v4i*)l,
        0, 0);
#elif USE_ASYNC_LDS
    const float* gf = (const float*)g;
    float* lf = (float*)l;
    #pragma unroll
    for (int k = 0; k < 4; ++k)
        __builtin_amdgcn_global_load_async_to_lds_b32(
            (__attribute__((address_space(1))) int*)(gf + k),
            (__attribute__((address_space(3))) int*)(lf + k),
            0, 0);
#else
    *l = *g;   // synchronous fallback
#endif
}

// ---------------------------------------------------------------------------
// Kernel 1: partial min/max of z = zm + exp(0.5*zv)*eps.
// grid = KB*KSPLIT blocks, 1024 threads. Lane owns 4 columns (float4) and a
// row phase (t>>8); each stage slot holds 4 eps rows (16 KB); depth-3 async
// pipeline keeps 3 slots of b128 copies in flight per wave.
// ---------------------------------------------------------------------------
__global__ __launch_bounds__(1024)
void k_minmax(const float* __restrict__ z_mean, const float* __restrict__ z_var,
              const float* __restrict__ eps,
              float* __restrict__ pmin, float* __restrict__ pmax) {
    __shared__ float4 stage[KSTG][1024];   // 48 KB
    __shared__ float red[64];

    const int blk   = blockIdx.x;
    const int b     = blk >> 3;            // KSPLIT == 8
    const int chunk = blk & (KSPLIT - 1);
    const int t     = threadIdx.x;
    const int wave  = t >> 5, lane = t & 31;
    const int tcol  = t & 255;             // float4 column index (0..255)
    const int trow  = t >> 8;              // row phase within a group (0..3)

    const float4 zm4 = ((const float4*)z_mean)[b * 256 + tcol];
    const float4 zv4 = ((const float4*)z_var)[b * 256 + tcol];
    float4 s4;
    s4.x = __expf(0.5f * zv4.x); s4.y = __expf(0.5f * zv4.y);
    s4.z = __expf(0.5f * zv4.z); s4.w = __expf(0.5f * zv4.w);

    // eps rows for this block, viewed as float4 (256 per row)
    const float4* ep4 = (const float4*)(eps + ((size_t)b * KMULT + (size_t)chunk * KROWS_PER_CHUNK) * KDIM);
    // source element for row-group g: row 4g+trow, col group tcol
    #define SRC(g) (ep4 + ((size_t)(4 * (g) + trow) << 8) + tcol)

    // Prologue: fill KSTG-1 slots.
    #pragma unroll
    for (int g = 0; g < KSTG - 1; ++g)
        stage_b128(SRC(g), &stage[g][t]);

    float mn =  __builtin_inff();
    float mx = -__builtin_inff();
    #pragma unroll
    for (int g = 0; g < KGROUPS; ++g) {
        int pf = g + KSTG - 1;
        if (pf > KGROUPS - 1) pf = KGROUPS - 1;       // clamped dup keeps counts uniform
        stage_b128(SRC(pf), &stage[(g + KSTG - 1) % KSTG][t]);
        wait_asynccnt<2 * KISS>();                    // oldest slot (group g) retired
        float4 e = stage[g % KSTG][t];
        float z0 = fmaf(s4.x, e.x, zm4.x);
        float z1 = fmaf(s4.y, e.y, zm4.y);
        float z2 = fmaf(s4.z, e.z, zm4.z);
        float z3 = fmaf(s4.w, e.w, zm4.w);
        mn = fminf(fminf(fminf(mn, z0), fminf(z1, z2)), z3);
        mx = fmaxf(fmaxf(fmaxf(mx, z0), fmaxf(z1, z2)), z3);
    }
    wait_asynccnt<0>();                               // drain
    #undef SRC

    // wave32 reduce, then cross-wave via LDS
    for (int o = 16; o > 0; o >>= 1) {
        mn = fminf(mn, __shfl_xor(mn, o, 32));
        mx = fmaxf(mx, __shfl_xor(mx, o, 32));
    }
    if (lane == 0) { red[wave] = mn; red[32 + wave] = mx; }
    __syncthreads();
    if (wave == 0) {
        float a = red[lane], c = red[32 + lane];
        for (int o = 16; o > 0; o >>= 1) {
            a = fminf(a, __shfl_xor(a, o, 32));
            c = fmaxf(c, __shfl_xor(c, o, 32));
        }
        if (lane == 0) { pmin[blk] = a; pmax[blk] = c; }
    }
}

// ---------------------------------------------------------------------------
// Kernel 2: per-row histogram into LDS (ds_add_u32), counts -> workspace.
// grid = KB blocks, 1024 threads. eps is L2-resident after kernel 1; b128 reads.
// ---------------------------------------------------------------------------
__global__ __launch_bounds__(1024)
void k_hist(const float* __restrict__ z_mean, const float* __restrict__ z_var,
            const float* __restrict__ eps,
            const float* __restrict__ pmin, const float* __restrict__ pmax,
            uint32_t* __restrict__ counts) {
    __shared__ uint32_t hist[KNB];         // 64 KB
    const int b    = blockIdx.x;
    const int t    = threadIdx.x;
    const int tcol = t & 255;
    const int trow = t >> 8;

    // Reduce the KSPLIT partials (redundant per thread; broadcast loads)
    float mn = pmin[b * KSPLIT], mx = pmax[b * KSPLIT];
    #pragma unroll
    for (int i = 1; i < KSPLIT; ++i) {
        mn = fminf(mn, pmin[b * KSPLIT + i]);
        mx = fmaxf(mx, pmax[b * KSPLIT + i]);
    }
    const float inv = (float)KNB / (mx - mn);

    const float4 zm4 = ((const float4*)z_mean)[b * 256 + tcol];
    const float4 zv4 = ((const float4*)z_var)[b * 256 + tcol];
    float4 s4;
    s4.x = __expf(0.5f * zv4.x); s4.y = __expf(0.5f * zv4.y);
    s4.z = __expf(0.5f * zv4.z); s4.w = __expf(0.5f * zv4.w);

    const float4* ep4 = (const float4*)(eps + (size_t)b * KMULT * KDIM);

    #pragma unroll
    for (int g = 0; g < 4; ++g) ((uint4*)hist)[4 * t + g] = make_uint4(0u, 0u, 0u, 0u);
    __syncthreads();

    for (int i = 0; i < KMULT / 4; ++i) {             // 32 iterations, 4 rows/iter
        const int row = 4 * i + trow;
        if (i + 4 < KMULT / 4)
            __builtin_prefetch(ep4 + (((size_t)row + 16) << 8) + tcol, 0, 1);  // global_prefetch
        float4 e = ep4[((size_t)row << 8) + tcol];
        float z0 = fmaf(s4.x, e.x, zm4.x);
        float z1 = fmaf(s4.y, e.y, zm4.y);
        float z2 = fmaf(s4.z, e.z, zm4.z);
        float z3 = fmaf(s4.w, e.w, zm4.w);
        int i0 = (int)((z0 - mn) * inv), i1 = (int)((z1 - mn) * inv);
        int i2 = (int)((z2 - mn) * inv), i3 = (int)((z3 - mn) * inv);
        i0 = i0 < 0 ? 0 : (i0 > KNB - 1 ? KNB - 1 : i0);
        i1 = i1 < 0 ? 0 : (i1 > KNB - 1 ? KNB - 1 : i1);
        i2 = i2 < 0 ? 0 : (i2 > KNB - 1 ? KNB - 1 : i2);
        i3 = i3 < 0 ? 0 : (i3 > KNB - 1 ? KNB - 1 : i3);
        atomicAdd(&hist[i0], 1u);                     // ds_add_u32
        atomicAdd(&hist[i1], 1u);
        atomicAdd(&hist[i2], 1u);
        atomicAdd(&hist[i3], 1u);
    }
    __syncthreads();

    uint4* crow = (uint4*)(counts + (size_t)b * KNB);
    #pragma unroll
    for (int g = 0; g < 4; ++g) crow[4 * t + g] = ((const uint4*)hist)[4 * t + g];
}

// ---------------------------------------------------------------------------
// Kernel 3: softmax over counts row + threshold + out = x * probs * MULT.
// grid = KB blocks, 1024 threads; thread owns 16 contiguous bins (4x uint4).
// ---------------------------------------------------------------------------
__global__ __launch_bounds__(1024)
void k_out(const uint32_t* __restrict__ counts, const float* __restrict__ x,
           float* __restrict__ out) {
    __shared__ float red[32];
    const int b = blockIdx.x, t = threadIdx.x;
    const int wave = t >> 5, lane = t & 31;

    const uint4*  crow = (const uint4*)(counts + (size_t)b * KNB);
    const float4* xrow = (const float4*)(x + (size_t)b * KNB);
    float4*       orow = (float4*)(out + (size_t)b * KNB);

    float c[16];
    #pragma unroll
    for (int g = 0; g < 4; ++g) {
        uint4 cc = crow[4 * t + g];
        c[4 * g + 0] = (float)cc.x; c[4 * g + 1] = (float)cc.y;
        c[4 * g + 2] = (float)cc.z; c[4 * g + 3] = (float)cc.w;
    }

    // --- row max ---
    float m = c[0];
    #pragma unroll
    for (int k = 1; k < 16; ++k) m = fmaxf(m, c[k]);
    for (int o = 16; o > 0; o >>= 1) m = fmaxf(m, __shfl_xor(m, o, 32));
    if (lane == 0) red[wave] = m;
    __syncthreads();
    if (wave == 0) {
        float v = red[lane];
        for (int o = 16; o > 0; o >>= 1) v = fmaxf(v, __shfl_xor(v, o, 32));
        if (lane == 0) red[0] = v;
    }
    __syncthreads();
    m = red[0];
    __syncthreads();                       // red about to be reused

    // --- sum of exp ---
    float e[16], ssum = 0.0f;
    #pragma unroll
    for (int k = 0; k < 16; ++k) { e[k] = __expf(c[k] - m); ssum += e[k]; }
    for (int o = 16; o > 0; o >>= 1) ssum += __shfl_xor(ssum, o, 32);
    if (lane == 0) red[wave] = ssum;
    __syncthreads();
    if (wave == 0) {
        float v = red[lane];
        for (int o = 16; o > 0; o >>= 1) v += __shfl_xor(v, o, 32);
        if (lane == 0) red[0] = v;
    }
    __syncthreads();
    const float invsum = 1.0f / red[0];

    #pragma unroll
    for (int g = 0; g < 4; ++g) {
        float4 xx = xrow[4 * t + g];
        float p0 = e[4 * g + 0] * invsum, p1 = e[4 * g + 1] * invsum;
        float p2 = e[4 * g + 2] * invsum, p3 = e[4 * g + 3] * invsum;
        p0 = (p0 < KZP) ? 0.0f : p0; p1 = (p1 < KZP) ? 0.0f : p1;
        p2 = (p2 < KZP) ? 0.0f : p2; p3 = (p3 < KZP) ? 0.0f : p3;
        float4 oo;
        oo.x = xx.x * p0 * (float)KMULT; oo.y = xx.y * p1 * (float)KMULT;
        oo.z = xx.z * p2 * (float)KMULT; oo.w = xx.w * p3 * (float)KMULT;
        orow[4 * t + g] = oo;
    }
}

// ---------------------------------------------------------------------------
extern "C" void kernel_launch(void* const* d_in, const int* in_sizes, int n_in,
                              void* d_out, int out_size, void* d_ws, size_t ws_size,
                              hipStream_t stream) {
    (void)in_sizes; (void)n_in; (void)out_size; (void)ws_size;
    const float* z_mean = (const float*)d_in[0];
    const float* z_var  = (const float*)d_in[1];
    const float* x      = (const float*)d_in[2];
    const float* eps    = (const float*)d_in[3];

    // Workspace layout: counts (128*16384 u32 = 8 MB) | pmin (4 KB) | pmax (4 KB)
    uint32_t* counts = (uint32_t*)d_ws;
    float* pmin = (float*)((char*)d_ws + (size_t)KB * KNB * sizeof(uint32_t));
    float* pmax = pmin + KB * KSPLIT;

    k_minmax<<<KB * KSPLIT, 1024, 0, stream>>>(z_mean, z_var, eps, pmin, pmax);
    k_hist  <<<KB,          1024, 0, stream>>>(z_mean, z_var, eps, pmin, pmax, counts);
    k_out   <<<KB,          1024, 0, stream>>>(counts, x, (float*)d_out);
}